// SimpleQuantumModel_88819923681730
// MI455X (gfx1250) — compile-verified
//
#include <hip/hip_runtime.h>

typedef float v2f __attribute__((ext_vector_type(2)));
typedef float v4f __attribute__((ext_vector_type(4)));
typedef float v8f __attribute__((ext_vector_type(8)));

#define VOCAB 40
#define ROWS_PER_WAVE 16
#define WAVES_PER_BLOCK 8
#define ROWS_PER_BLOCK (ROWS_PER_WAVE * WAVES_PER_BLOCK)   // 128

// out[i, v] = bias[v] + sum_{k=1..4} W[v, k-1] * cos(x[i])^k
// (q_weights cancel: |exp(i*phi)| == 1, so probs = state^2 and
//  expvals reduce analytically to [u, u^2, u^3, u^4], u = cos(x).)
// Computed as 16x16x4 fp32 WMMA: A = u-powers per row, B = W^T tile, C = bias.
__global__ __launch_bounds__(256) void quantum_poly_wmma(
    const float* __restrict__ x,
    const float* __restrict__ fc_weight,   // (40, 4) row-major
    const float* __restrict__ fc_bias,     // (40,)
    float* __restrict__ out,               // (SEQ, 40) row-major
    int seq)
{
    __shared__ float lds[ROWS_PER_BLOCK * VOCAB];          // 20 KB / block

    const int tid  = threadIdx.x;
    const int wave = tid >> 5;
    const int lane = tid & 31;
    const int half = lane >> 4;        // 0: lanes 0-15 (K=0,1), 1: lanes 16-31 (K=2,3)
    const int l16  = lane & 15;

    const int row_base = blockIdx.x * ROWS_PER_BLOCK + wave * ROWS_PER_WAVE;

    // ---- A fragment: 16x4 f32 (ISA 7.12.2). Lane holds M = lane%16,
    //      VGPR0 = K(2*half), VGPR1 = K(2*half+1); A[m][k] = u^(k+1). ----
    const float theta = x[row_base + l16];
    const float u  = cosf(theta);
    const float u2 = u * u;
    v2f a;
    a.x = half ? (u * u2)  : u;    // u^3 : u^1
    a.y = half ? (u2 * u2) : u2;   // u^4 : u^2

    // ---- 3 N-tiles covering vocab columns [0,16), [16,32), [32,48) ----
    #pragma unroll
    for (int t = 0; t < 3; ++t) {
        const int  ncol   = t * 16 + l16;
        const bool nvalid = (ncol < VOCAB);
        const int  nclamp = nvalid ? ncol : (VOCAB - 1);   // keep loads in-bounds

        // B fragment: 4x16 f32, B[k][n] = fc_weight[n][k].
        // Lane holds N = lane%16; VGPR0 = K(2*half), VGPR1 = K(2*half+1).
        // Direct lane-addressed load -> single global_load_b64, no cndmask chain.
        const float* wrow = fc_weight + nclamp * 4 + 2 * half;
        v2f b;
        b.x = nvalid ? wrow[0] : 0.0f;
        b.y = nvalid ? wrow[1] : 0.0f;
        const float bias = nvalid ? fc_bias[nclamp] : 0.0f;

        // C fragment: bias broadcast down each column (independent of M).
        v8f c;
        #pragma unroll
        for (int r = 0; r < 8; ++r) c[r] = bias;

        // D = A x B + C  ->  v_wmma_f32_16x16x4_f32
        v8f d = __builtin_amdgcn_wmma_f32_16x16x4_f32(
            /*neg_a=*/false, a, /*neg_b=*/false, b,
            /*c_mod=*/(short)0, c, /*reuse_a=*/false, /*reuse_b=*/false);

        // D layout: VGPR r holds M = r + 8*half, N = lane%16.
        if (nvalid) {
            const int mbase = wave * ROWS_PER_WAVE + 8 * half;
            #pragma unroll
            for (int r = 0; r < 8; ++r)
                lds[(mbase + r) * VOCAB + ncol] = d[r];
        }
    }

    __syncthreads();

    // ---- Coalesced streaming store: block owns a contiguous 5120-float slice ----
    const v4f* lds4 = (const v4f*)lds;
    v4f* out4 = (v4f*)out;
    const int base4 = blockIdx.x * (ROWS_PER_BLOCK * VOCAB / 4);   // 1280 per block
    #pragma unroll
    for (int j = 0; j < (ROWS_PER_BLOCK * VOCAB / 4) / 256; ++j) { // 5 iters
        out4[base4 + j * 256 + tid] = lds4[j * 256 + tid];
    }
    (void)seq;
}

extern "C" void kernel_launch(void* const* d_in, const int* in_sizes, int n_in,
                              void* d_out, int out_size, void* d_ws, size_t ws_size,
                              hipStream_t stream) {
    // Inputs (setup_inputs order): x (1,SEQ) f32; q_weights (4,) f32 [unused:
    // unit-magnitude phase cancels in |psi|^2]; fc_weight (40,4) f32;
    // fc_bias (40,) f32. Output: (SEQ, 40) f32.
    const float* x    = (const float*)d_in[0];
    const float* fc_w = (const float*)d_in[2];
    const float* fc_b = (const float*)d_in[3];
    float* out = (float*)d_out;

    const int seq = in_sizes[0];                 // 1048576
    const int blocks = seq / ROWS_PER_BLOCK;     // 8192
    hipLaunchKernelGGL(quantum_poly_wmma, dim3(blocks), dim3(256), 0, stream,
                       x, fc_w, fc_b, out, seq);
    (void)n_in; (void)out_size; (void)d_ws; (void)ws_size; (void)in_sizes;
}